// GritLMPooler_53480932770475
// MI455X (gfx1250) — compile-verified
//
#include <hip/hip_runtime.h>
#include <math.h>
#include <stdint.h>

// GritLM pooler: masked mean over ragged segments + L2 normalize.
// Bandwidth-bound: ~1.2 GB streamed @ 23.3 TB/s => ~51 us floor.
// Phase 1: V_WMMA_F32_16X16X4_F32 with A=ones as a full-f32 column summer
// (16 cols per wave). Main loop is branchless, 16 tokens/iter: 8 independent
// global_load_b32 with immediate offsets, then 4 WMMAs on 2 accumulators.
// Phase 2 combines partials, computes mean and L2-normalizes.

#define NSEQ        32
#define DIM         4096
#define COLS_BLK    128                // columns per block (8 waves x 16 cols)
#define CHUNKS      (DIM / COLS_BLK)   // 32
#define RSPLIT      2                  // row-range split per sequence
#define EPSN        1e-12f

typedef __attribute__((ext_vector_type(2))) float v2f;
typedef __attribute__((ext_vector_type(8))) float v8f;

__device__ __forceinline__ v8f wmma_sum4(v2f a, v2f b, v8f c) {
    // (neg_a, A, neg_b, B, c_mod, C, reuse_a, reuse_b)
    return __builtin_amdgcn_wmma_f32_16x16x4_f32(false, a, false, b,
                                                 (short)0, c, false, false);
}

__global__ __launch_bounds__(256) void grit_pool_phase1(
    const float* __restrict__ H,      // [T, DIM]
    const int*   __restrict__ plen,   // [NSEQ]
    const int*   __restrict__ ilen,   // [NSEQ]
    float*       __restrict__ part)   // [NSEQ, RSPLIT, DIM] column sums
{
    const int bid   = blockIdx.x;
    const int seq   = bid / (CHUNKS * RSPLIT);
    const int rem   = bid % (CHUNKS * RSPLIT);
    const int chunk = rem / RSPLIT;
    const int rs    = rem % RSPLIT;

    // Uniform (scalar-path) prefix sum over 32 lengths to get segment start.
    int cu = 0;
    for (int i = 0; i < seq; ++i) cu += plen[i];
    const int nvalid = plen[seq] - ilen[seq];     // >= 1 by construction
    const int start  = cu + ilen[seq];
    const int nhalf  = nvalid / 2;                // RSPLIT == 2
    const long rbeg  = start + (rs ? nhalf : 0);
    const long rend  = start + (rs ? nvalid : nhalf);

    const int wave  = threadIdx.x >> 5;           // 0..7
    const int lane  = threadIdx.x & 31;
    const int n_idx = lane & 15;                  // column within wave tile
    const int k_off = (lane >> 4) << 1;           // 0 (lanes 0-15) / 2 (16-31)

    const int col = chunk * COLS_BLK + wave * 16 + n_idx;
    const float* __restrict__ bp = H + col;

    // A = 16x4 all-ones -> D[m,n] = sum_k B[k,n]; accumulate C across steps.
    const v2f a = {1.0f, 1.0f};
    v8f acc0 = {0.f, 0.f, 0.f, 0.f, 0.f, 0.f, 0.f, 0.f};
    v8f acc1 = {0.f, 0.f, 0.f, 0.f, 0.f, 0.f, 0.f, 0.f};

    // ---- Main loop: 16 tokens/iter, no guards, 8 loads then 4 WMMAs ----
    const long nfull = (rend > rbeg) ? (rend - rbeg) / 16 : 0;
    const float* p = bp + (rbeg + (long)k_off) * (long)DIM;
    for (long i = 0; i < nfull; ++i) {
        // Constant indices -> immediate byte offsets (max 13*16384 < 2^23).
        const float b0 = p[0L * DIM];
        const float b1 = p[1L * DIM];
        const float b2 = p[4L * DIM];
        const float b3 = p[5L * DIM];
        const float b4 = p[8L * DIM];
        const float b5 = p[9L * DIM];
        const float b6 = p[12L * DIM];
        const float b7 = p[13L * DIM];
        // 16KB row stride defeats HW prefetch; pull ~64 rows (1MB) ahead.
        __builtin_prefetch(p + 64L * DIM, 0, 1);
        v2f bA = {b0, b1};
        v2f bB = {b2, b3};
        v2f bC = {b4, b5};
        v2f bD = {b6, b7};
        acc0 = wmma_sum4(a, bA, acc0);
        acc1 = wmma_sum4(a, bB, acc1);
        acc0 = wmma_sum4(a, bC, acc0);
        acc1 = wmma_sum4(a, bD, acc1);
        p += 16L * DIM;
    }

    // ---- Tail: up to 15 tokens, guarded 4-token steps ----
    for (long r = rbeg + nfull * 16; r < rend; r += 4) {
        const long t0 = r + k_off;
        const long t1 = t0 + 1;
        const float b0 = (t0 < rend) ? bp[t0 * (long)DIM] : 0.0f;
        const float b1 = (t1 < rend) ? bp[t1 * (long)DIM] : 0.0f;
        v2f b = {b0, b1};
        acc0 = wmma_sum4(a, b, acc0);
    }

    // All 16 output rows are identical; element 0 lanes 0-15 = cols 0..15 (M=0).
    if (lane < 16) {
        part[(size_t)(seq * RSPLIT + rs) * DIM + col] = acc0[0] + acc1[0];
    }
}

__global__ __launch_bounds__(128) void grit_pool_phase2(
    const float* __restrict__ part,   // [NSEQ, RSPLIT, DIM]
    const int*   __restrict__ plen,
    const int*   __restrict__ ilen,
    float*       __restrict__ out)    // [NSEQ, DIM]
{
    const int seq = blockIdx.x;
    const int tid = threadIdx.x;
    const float inv_n = 1.0f / (float)(plen[seq] - ilen[seq]);

    const float4* __restrict__ p0 =
        (const float4*)(part + (size_t)(seq * RSPLIT + 0) * DIM);
    const float4* __restrict__ p1 =
        (const float4*)(part + (size_t)(seq * RSPLIT + 1) * DIM);

    float4 m[8];
    float ssq = 0.0f;
#pragma unroll
    for (int i = 0; i < 8; ++i) {
        const int g = tid + i * 128;              // float4 group: 1024 total
        const float4 x = p0[g];
        const float4 y = p1[g];
        float4 mm;
        mm.x = (x.x + y.x) * inv_n;
        mm.y = (x.y + y.y) * inv_n;
        mm.z = (x.z + y.z) * inv_n;
        mm.w = (x.w + y.w) * inv_n;
        ssq += mm.x * mm.x + mm.y * mm.y + mm.z * mm.z + mm.w * mm.w;
        m[i] = mm;
    }

    __shared__ float red[128];
    red[tid] = ssq;
    __syncthreads();
    for (int s = 64; s > 0; s >>= 1) {
        if (tid < s) red[tid] += red[tid + s];
        __syncthreads();
    }
    const float inv = 1.0f / fmaxf(sqrtf(red[0]), EPSN);

    float4* __restrict__ o = (float4*)(out + (size_t)seq * DIM);
#pragma unroll
    for (int i = 0; i < 8; ++i) {
        float4 mm = m[i];
        mm.x *= inv; mm.y *= inv; mm.z *= inv; mm.w *= inv;
        o[tid + i * 128] = mm;
    }
}

extern "C" void kernel_launch(void* const* d_in, const int* in_sizes, int n_in,
                              void* d_out, int out_size, void* d_ws, size_t ws_size,
                              hipStream_t stream) {
    const float* H    = (const float*)d_in[0];   // [T, 4096] f32
    const int*   plen = (const int*)d_in[1];     // [32]
    const int*   ilen = (const int*)d_in[2];     // [32]
    float*       outp = (float*)d_out;           // [32, 4096] f32

    // Scratch: NSEQ * RSPLIT * DIM floats = 1 MiB of partial column sums.
    float* part = (float*)d_ws;

    grit_pool_phase1<<<NSEQ * CHUNKS * RSPLIT, 256, 0, stream>>>(H, plen, ilen, part);
    grit_pool_phase2<<<NSEQ, 128, 0, stream>>>(part, plen, ilen, outp);
    (void)in_sizes; (void)n_in; (void)out_size; (void)ws_size;
}